// Gate_28905129902147
// MI455X (gfx1250) — compile-verified
//
#include <hip/hip_runtime.h>
#include <hip/hip_bf16.h>

typedef float v2f __attribute__((ext_vector_type(2)));
typedef float v4f __attribute__((ext_vector_type(4)));
typedef float v8f __attribute__((ext_vector_type(8)));

// ---------------------------------------------------------------------------
// Kernel 1: global average pool. One wave32 per (b,c) row of 1024 floats.
// 192 MiB streamed once -> NT loads, b128 vectorized.
// ---------------------------------------------------------------------------
__global__ __launch_bounds__(256) void pool_kernel(const float* __restrict__ x,
                                                   float* __restrict__ pooled,
                                                   int nrows) {
    const int wavesPerBlock = blockDim.x >> 5;
    const int row  = blockIdx.x * wavesPerBlock + (threadIdx.x >> 5);
    const int lane = threadIdx.x & 31;
    if (row >= nrows) return;

    const v4f* p = reinterpret_cast<const v4f*>(x + (size_t)row * 1024);
    float s = 0.0f;
#pragma unroll
    for (int i = 0; i < 8; ++i) {
        v4f v = __builtin_nontemporal_load(&p[lane + i * 32]);
        s += v.x + v.y + v.z + v.w;
    }
#pragma unroll
    for (int off = 16; off > 0; off >>= 1)
        s += __shfl_xor(s, off, 32);

    if (lane == 0) pooled[row] = s * (1.0f / 1024.0f);
}

// ---------------------------------------------------------------------------
// Kernel 2: scores = sigmoid(pooled @ W^T + b) via v_wmma_f32_16x16x4_f32.
// One 512-thread block; 16 waves; wave (mi,ni) owns a 16x16 tile of the
// 64x64 result. K=768 consumed 4 at a time (f32 WMMA shape 16x16x4).
// ---------------------------------------------------------------------------
__global__ __launch_bounds__(512) void gemm_sigmoid_kernel(const float* __restrict__ pooled,
                                                           const float* __restrict__ Wm,
                                                           const float* __restrict__ bias,
                                                           float* __restrict__ scores) {
    const int lane    = threadIdx.x & 31;
    const int wave    = threadIdx.x >> 5;
    const int mi      = wave >> 2;        // M tile 0..3
    const int ni      = wave & 3;         // N tile 0..3
    const int halfsel = lane >> 4;        // 0: K pair {0,1}, 1: K pair {2,3}
    const int mrow    = mi * 16 + (lane & 15);   // A: lane carries row M
    const int ecol    = ni * 16 + (lane & 15);   // B: lane carries col N (expert)

    v8f c = {};
    for (int k = 0; k < 768; k += 4) {
        const int kk = k + (halfsel << 1);
        v2f a, b;
        // A 16x4 layout: VGPR0 = K=kk, VGPR1 = K=kk+1 (per ISA table)
        a.x = pooled[mrow * 768 + kk];
        a.y = pooled[mrow * 768 + kk + 1];
        // B 4x16 layout: B[K][N] = W^T[K][N] = Wm[N*768 + K]
        b.x = Wm[ecol * 768 + kk];
        b.y = Wm[ecol * 768 + kk + 1];
        // 8 args: (neg_a, A, neg_b, B, c_mod, C, reuse_a, reuse_b)
        c = __builtin_amdgcn_wmma_f32_16x16x4_f32(false, a, false, b,
                                                  (short)0, c, false, false);
    }

    const float be = bias[ecol];
#pragma unroll
    for (int r = 0; r < 8; ++r) {
        // C/D 16x16 layout: VGPR r holds M=r (lanes 0-15) / M=r+8 (lanes 16-31)
        const int m = mi * 16 + r + (halfsel << 3);
        float v = c[r] + be;
        v = 1.0f / (1.0f + __expf(-v));
        scores[m * 64 + ecol] = v;
    }
}

// ---------------------------------------------------------------------------
// Kernel 3: per-row top-8 on (scores + route_bias), gather original scores,
// normalize. One wave32 per batch row; 2 candidates per lane; 8 rounds of
// shuffle argmax with lower-index tie-break (matches lax.top_k ordering).
// ---------------------------------------------------------------------------
__global__ __launch_bounds__(256) void topk_kernel(const float* __restrict__ scores,
                                                   const float* __restrict__ route_bias,
                                                   float* __restrict__ outW,
                                                   int* __restrict__ outI) {
    const int row  = blockIdx.x * (blockDim.x >> 5) + (threadIdx.x >> 5);
    const int lane = threadIdx.x & 31;

    float s0 = scores[row * 64 + lane];
    float s1 = scores[row * 64 + lane + 32];
    float b0 = s0 + route_bias[lane];
    float b1 = s1 + route_bias[lane + 32];

    float wsel = 0.0f;
    int   isel = 0;
    float total = 0.0f;

#pragma unroll
    for (int t = 0; t < 8; ++t) {
        float v;
        int   idx;
        if (b1 > b0) { v = b1; idx = lane + 32; }
        else         { v = b0; idx = lane; }
#pragma unroll
        for (int off = 16; off > 0; off >>= 1) {
            float ov = __shfl_xor(v, off, 32);
            int   oi = __shfl_xor(idx, off, 32);
            if (ov > v || (ov == v && oi < idx)) { v = ov; idx = oi; }
        }
        // idx is now wave-uniform: fetch the ORIGINAL (unbiased) score
        float orig = __shfl(idx < 32 ? s0 : s1, idx & 31, 32);
        total += orig;
        if (lane == t) { wsel = orig; isel = idx; }
        // mask the winner out of future rounds
        if ((idx & 31) == lane) {
            if (idx < 32) b0 = -3.0e38f; else b1 = -3.0e38f;
        }
    }

    if (lane < 8) {
        outW[row * 8 + lane] = wsel / total;   // ROUTE_SCALE == 1.0
        outI[row * 8 + lane] = isel;
    }
}

// ---------------------------------------------------------------------------
extern "C" void kernel_launch(void* const* d_in, const int* in_sizes, int n_in,
                              void* d_out, int out_size, void* d_ws, size_t ws_size,
                              hipStream_t stream) {
    const float* x          = (const float*)d_in[0];  // [64,768,32,32]
    const float* Wm         = (const float*)d_in[1];  // [64,768]
    const float* bias       = (const float*)d_in[2];  // [64]
    const float* route_bias = (const float*)d_in[3];  // [64]
    // d_in[4] = topk scalar (constant 8 in reference)

    const int B = 64, C = 768, E = 64, K = 8;
    const int nrows = B * C;                          // 49152 pooled rows

    float* pooled = (float*)d_ws;                     // 64*768 floats = 192 KiB
    float* scores = pooled + (size_t)nrows;           // 64*64 floats  =  16 KiB

    float* outW = (float*)d_out;                      // [64,8] weights
    int*   outI = (int*)(outW + B * K);               // [64,8] indices (int32)

    // 1) pooling: 8 waves per block -> 8 rows per block
    pool_kernel<<<nrows / 8, 256, 0, stream>>>(x, pooled, nrows);

    // 2) 64x64 GEMM + bias + sigmoid in one workgroup (16 waves, WMMA f32)
    gemm_sigmoid_kernel<<<1, 512, 0, stream>>>(pooled, Wm, bias, scores);

    // 3) top-8 + normalize: 64 rows, 8 waves per block -> 8 blocks
    topk_kernel<<<B / 8, 256, 0, stream>>>(scores, route_bias, outW, outI);
}